// PermutationClosedLayer_53145925321282
// MI455X (gfx1250) — compile-verified
//
#include <hip/hip_runtime.h>

// CDNA5 / gfx1250 wave32 WMMA kernel for PermutationClosedLayer.
//
// out[s,r,o] = pooled0[s,r,:] . (W0-W1)[o,:]  +  total[s,:] . W1[o,:]
//   pooled0[s,r,c] = SA[maskA(r)][c] + SB[maskB(r)][c]   (5-subset sums via LDS DP)
//   base[s,o]      = total[s,:] . W1[o,:]   -> folded into the WMMA C accumulator.
//   Wd = W0 - W1 precomputed once per block into LDS; B fragments stay in VGPRs.
//   Last r-tile (rows 240..255, only 240..251 valid) peeled so the hot loop has
//   unconditional stores.
//
// Uses V_WMMA_F32_16X16X4_F32 (full fp32; kernel is bound by the 132 MB output
// stream, so there is no reason to drop precision).

typedef __attribute__((ext_vector_type(2))) float v2f;
typedef __attribute__((ext_vector_type(8))) float v8f;

#define SIZEP 10
#define CIN   32
#define COUT  32
#define NR    252
#define WAVES 4   // waves per block; one sample per wave

__device__ __forceinline__ void wmma_rtile(const v2f* pSA, const v2f* pSB,
                                           const v2f b0[8], const v2f b1[8],
                                           int halfsel, float bz0, float bz1,
                                           v8f& acc0, v8f& acc1) {
  acc0 = (v8f){bz0, bz0, bz0, bz0, bz0, bz0, bz0, bz0};
  acc1 = (v8f){bz1, bz1, bz1, bz1, bz1, bz1, bz1, bz1};
  #pragma unroll
  for (int kk = 0; kk < 8; ++kk) {
    const int ch = 2 * kk + halfsel;               // float2 index of K-pair
    const v2f a  = pSA[ch] + pSB[ch];              // pooled0 fragment
    // 8 args: (neg_a, A, neg_b, B, c_mod, C, reuse_a, reuse_b)
    acc0 = __builtin_amdgcn_wmma_f32_16x16x4_f32(
        false, a, false, b0[kk], (short)0, acc0, false, false);
    acc1 = __builtin_amdgcn_wmma_f32_16x16x4_f32(
        false, a, false, b1[kk], (short)0, acc1, false, false);
  }
}

__global__ __launch_bounds__(WAVES * 32)
void pcl_wmma_kernel(const float* __restrict__ x,     // [samples][10][32]
                     const float* __restrict__ W,     // [2][32][32]
                     const int*   __restrict__ split0,// [252][5] (first half of splits)
                     float* __restrict__ out,         // [samples][252][32]
                     int samples) {
  __shared__ __align__(16) unsigned int s_mask[256];           // maskA | maskB<<8
  __shared__ __align__(16) float s_Wd[COUT * CIN];             // W0 - W1, [o][c]
  __shared__ __align__(16) float s_SA[WAVES][32 * 32];         // subset sums, pos 0..4
  __shared__ __align__(16) float s_SB[WAVES][32 * 32];         // subset sums, pos 5..9
  __shared__ __align__(16) float s_base[WAVES][32];            // total . W1^T

  const int tid = threadIdx.x;
  const int w   = tid >> 5;        // wave id in block
  const int l   = tid & 31;        // lane id (wave32)
  const int s   = blockIdx.x * WAVES + w;

  // ---- split masks (shared by all waves) -------------------------------
  for (int r = tid; r < 256; r += WAVES * 32) {
    unsigned int mA = 0u, mB = 0u;
    if (r < NR) {
      #pragma unroll
      for (int g = 0; g < 5; ++g) {
        int p = split0[r * 5 + g];
        if (p < 5) mA |= 1u << p;
        else       mB |= 1u << (p - 5);
      }
    }
    s_mask[r] = mA | (mB << 8);
  }

  // ---- Wd = W0 - W1 into LDS (shared by all waves) ---------------------
  #pragma unroll
  for (int i = 0; i < (COUT * CIN) / (WAVES * 32); ++i) {
    const int idx = tid + i * (WAVES * 32);
    s_Wd[idx] = W[idx] - W[COUT * CIN + idx];
  }

  // ---- per-sample subset sums: lane == channel -------------------------
  if (s < samples) {
    const float* xs = x + (size_t)s * (SIZEP * CIN);
    float xa[5], xb[5];
    #pragma unroll
    for (int p = 0; p < 5; ++p) {
      xa[p] = xs[p * CIN + l];
      xb[p] = xs[(p + 5) * CIN + l];
    }
    float sa[32], sb[32];
    sa[0] = 0.f; sb[0] = 0.f;
    #pragma unroll
    for (int m = 1; m < 32; ++m) {
      sa[m] = sa[m & (m - 1)] + xa[__builtin_ctz(m)];
      sb[m] = sb[m & (m - 1)] + xb[__builtin_ctz(m)];
    }
    #pragma unroll
    for (int m = 0; m < 32; ++m) {
      s_SA[w][m * 32 + l] = sa[m];
      s_SB[w][m * 32 + l] = sb[m];
    }
  }
  __syncthreads();

  // ---- base[s,o] = total . W1[o,:]  (lane == output channel o) ---------
  if (s < samples) {
    const float* w1row = W + COUT * CIN + l * CIN;   // W[1][l][:]
    float acc = 0.f;
    #pragma unroll
    for (int c = 0; c < CIN; ++c)
      acc += (s_SA[w][31 * 32 + c] + s_SB[w][31 * 32 + c]) * w1row[c];
    s_base[w][l] = acc;
  }
  __syncthreads();

  if (s >= samples) return;

  const int halfsel = l >> 4;   // 0: K-pair {4k,4k+1}, 1: {4k+2,4k+3}
  const int nlo     = l & 15;   // N (cout) / M (row) low index

  // ---- B fragments: load once from LDS into VGPRs ----------------------
  // lane l holds Wd[o = nt*16+nlo][4*kk + 2*halfsel + {0,1}]
  v2f b0[8], b1[8];
  {
    const v2f* pB0 = (const v2f*)&s_Wd[nlo * CIN];
    const v2f* pB1 = (const v2f*)&s_Wd[(16 + nlo) * CIN];
    #pragma unroll
    for (int kk = 0; kk < 8; ++kk) {
      b0[kk] = pB0[2 * kk + halfsel];
      b1[kk] = pB1[2 * kk + halfsel];
    }
  }

  const float bz0 = s_base[w][nlo];
  const float bz1 = s_base[w][16 + nlo];

  float* outp = out + (size_t)s * (NR * COUT);

  // ---- hot loop: r-tiles 0..14, all 16 rows valid, unconditional stores -
  for (int rt = 0; rt < 15; ++rt) {
    const unsigned int mm = s_mask[rt * 16 + nlo];
    const v2f* pSA = (const v2f*)&s_SA[w][(mm & 31u) * 32];
    const v2f* pSB = (const v2f*)&s_SB[w][((mm >> 8) & 31u) * 32];

    v8f acc0, acc1;
    wmma_rtile(pSA, pSB, b0, b1, halfsel, bz0, bz1, acc0, acc1);

    // D layout: VGPR j, lane l -> row M = j + 8*halfsel, col N = nlo
    float* orow = outp + (rt * 16 + 8 * halfsel) * COUT + nlo;
    #pragma unroll
    for (int j = 0; j < 8; ++j) {
      orow[j * COUT]      = acc0[j];
      orow[j * COUT + 16] = acc1[j];
    }
  }

  // ---- peeled last r-tile: rows 240..255, store only 240..251 ----------
  {
    const unsigned int mm = s_mask[15 * 16 + nlo];
    const v2f* pSA = (const v2f*)&s_SA[w][(mm & 31u) * 32];
    const v2f* pSB = (const v2f*)&s_SB[w][((mm >> 8) & 31u) * 32];

    v8f acc0, acc1;
    wmma_rtile(pSA, pSB, b0, b1, halfsel, bz0, bz1, acc0, acc1);

    float* orow = outp + (15 * 16 + 8 * halfsel) * COUT + nlo;
    #pragma unroll
    for (int j = 0; j < 8; ++j) {
      if (8 * halfsel + j < NR - 240) {   // j + 8*halfsel < 12
        orow[j * COUT]      = acc0[j];
        orow[j * COUT + 16] = acc1[j];
      }
    }
  }
}

extern "C" void kernel_launch(void* const* d_in, const int* in_sizes, int n_in,
                              void* d_out, int out_size, void* d_ws, size_t ws_size,
                              hipStream_t stream) {
  const float* x      = (const float*)d_in[0];
  const float* W      = (const float*)d_in[1];
  const int*   splits = (const int*)d_in[2];   // [2][252][5]; we use split0 half
  float* out = (float*)d_out;

  const int samples = in_sizes[0] / (SIZEP * CIN);
  const int blocks  = (samples + WAVES - 1) / WAVES;

  pcl_wmma_kernel<<<blocks, WAVES * 32, 0, stream>>>(x, W, splits, out, samples);
}